// MemoryBank_21294447854175
// MI455X (gfx1250) — compile-verified
//
#include <hip/hip_runtime.h>
#include <math.h>

// ----------------------------------------------------------------------------
// MemoryBank update for MI455X (gfx1250, wave32)
//   out = features_memory, then for each sample i in order:
//       out[y_i] = normalize(0.01*out[y_i] + 0.99*x_i)
// Phase 1: bulk copy feat -> out (bandwidth bound, ~820MB traffic) via the
//          CDNA5 async global<->LDS DMA path (GLOBAL_LOAD_ASYNC_TO_LDS_B128 /
//          GLOBAL_STORE_ASYNC_FROM_LDS_B128, ASYNCcnt).
// Phase 2: one wave32 block per sample; only the first occurrence of a target
//          processes the full collision chain sequentially (exact semantics).
// ----------------------------------------------------------------------------

typedef float v4f __attribute__((ext_vector_type(4)));
typedef int   v4i __attribute__((vector_size(16)));              // matches builtin param
typedef __attribute__((address_space(1))) v4i* global_v4i_p;     // __device__ int4*
typedef __attribute__((address_space(3))) v4i* lds_v4i_p;        // __shared__ int4*

#define MOM 0.01f
#define THREADS_COPY 256
#define K_UNROLL 4
#define TILE_ELEMS (THREADS_COPY * K_UNROLL)   // float4 elements per block tile

#if defined(__AMDGCN__) &&                                                    \
    __has_builtin(__builtin_amdgcn_global_load_async_to_lds_b128) &&          \
    __has_builtin(__builtin_amdgcn_global_store_async_from_lds_b128) &&       \
    __has_builtin(__builtin_amdgcn_s_wait_asynccnt)
#define USE_ASYNC_LDS 1
#else
#define USE_ASYNC_LDS 0
#endif

// ---------------- Phase 1: bulk copy ----------------------------------------
__global__ __launch_bounds__(THREADS_COPY)
void mb_copy(const v4f* __restrict__ src, v4f* __restrict__ dst, long long n16)
{
#if USE_ASYNC_LDS
    // Global -> LDS -> Global without touching VGPRs (ASYNCcnt tracked).
    __shared__ char lbuf[TILE_ELEMS * 16];
    const int tid = threadIdx.x;
    const long long ntiles = (n16 + TILE_ELEMS - 1) / TILE_ELEMS;
    for (long long t = blockIdx.x; t < ntiles; t += gridDim.x) {
        const long long base = t * (long long)TILE_ELEMS + tid;
#pragma unroll
        for (int k = 0; k < K_UNROLL; ++k) {
            const long long idx = base + (long long)k * THREADS_COPY;
            if (idx < n16) {
                __builtin_amdgcn_global_load_async_to_lds_b128(
                    (global_v4i_p)(src + idx),
                    (lds_v4i_p)&lbuf[(k * THREADS_COPY + tid) * 16],
                    0, 0);
            }
        }
        __builtin_amdgcn_s_wait_asynccnt(0);   // loads landed in LDS
#pragma unroll
        for (int k = 0; k < K_UNROLL; ++k) {
            const long long idx = base + (long long)k * THREADS_COPY;
            if (idx < n16) {
                __builtin_amdgcn_global_store_async_from_lds_b128(
                    (global_v4i_p)(dst + idx),
                    (lds_v4i_p)&lbuf[(k * THREADS_COPY + tid) * 16],
                    0, 0);
            }
        }
        __builtin_amdgcn_s_wait_asynccnt(0);   // stores drained, LDS reusable
    }
#else
    // Fallback: plain grid-stride streaming B128 copy with NT hints.
    long long i = (long long)blockIdx.x * THREADS_COPY + threadIdx.x;
    const long long stride = (long long)gridDim.x * THREADS_COPY;
    for (; i < n16; i += stride) {
        v4f v = __builtin_nontemporal_load(&src[i]);
        __builtin_nontemporal_store(v, &dst[i]);
    }
#endif
}

// ---------------- Phase 2: sequential-chain momentum update -----------------
// One wave32 block per sample. D = 256 floats: 8 floats (2 x float4) per lane.
__global__ __launch_bounds__(32)
void mb_update(const float* __restrict__ x,     // [B, 256]
               const int*   __restrict__ tgt,   // [B]
               const float* __restrict__ feat,  // [C, 256] original (input)
               float*       __restrict__ out,   // [C, 256] result
               int B)
{
    const int i    = blockIdx.x;
    const int lane = threadIdx.x;
    const int y    = tgt[i];

    // Is sample i the first occurrence of target y?  (lanes scan tgt[0..i))
    int found = 0;
    for (int j = lane; j < i; j += 32) {
        if (tgt[j] == y) { found = 1; break; }
    }
    if (__ballot(found) != 0ULL) return;   // a predecessor owns this chain

    // Chain start: original row y from the (immutable) input tensor.
    const v4f* frow = (const v4f*)(feat + (long long)y * 256);
    v4f a0 = frow[lane * 2 + 0];
    v4f a1 = frow[lane * 2 + 1];

    // Walk all occurrences of y in sample order (usually just one).
    for (int j = i; j < B; ++j) {
        if (__builtin_expect(tgt[j] != y, 1)) continue;  // uniform scalar test
        const v4f* xr = (const v4f*)(x + (long long)j * 256);
        __builtin_prefetch(xr, 0, 1);
        v4f b0 = xr[lane * 2 + 0];
        v4f b1 = xr[lane * 2 + 1];
        a0 = MOM * a0 + (1.0f - MOM) * b0;
        a1 = MOM * a1 + (1.0f - MOM) * b1;
        // ||row||^2 : 8 local squares, then wave32 butterfly reduction.
        float s = a0.x * a0.x + a0.y * a0.y + a0.z * a0.z + a0.w * a0.w +
                  a1.x * a1.x + a1.y * a1.y + a1.z * a1.z + a1.w * a1.w;
        for (int off = 16; off > 0; off >>= 1)
            s += __shfl_xor(s, off, 32);
        const float inv = 1.0f / sqrtf(s);
        a0 *= inv;
        a1 *= inv;
    }

    v4f* orow = (v4f*)(out + (long long)y * 256);
    orow[lane * 2 + 0] = a0;
    orow[lane * 2 + 1] = a1;
}

// ---------------- host entry -------------------------------------------------
extern "C" void kernel_launch(void* const* d_in, const int* in_sizes, int n_in,
                              void* d_out, int out_size, void* d_ws, size_t ws_size,
                              hipStream_t stream)
{
    const float* x    = (const float*)d_in[0];   // inputs          [B, D] f32
    const int*   tgt  = (const int*)  d_in[1];   // targets         [B]
    const float* feat = (const float*)d_in[2];   // features_memory [C, D] f32
    float*       out  = (float*)d_out;           // [C, D] f32

    const int B = in_sizes[1];
    const long long n   = (long long)in_sizes[2];   // C * D floats (D == 256)
    const long long n16 = n / 4;                    // float4 elements

    const long long ntiles = (n16 + TILE_ELEMS - 1) / TILE_ELEMS;
    const int grid_copy = (int)(ntiles < 16384 ? ntiles : 16384);

    // Phase 1 then Phase 2 — same stream, so ordering is guaranteed.
    mb_copy<<<grid_copy, THREADS_COPY, 0, stream>>>((const v4f*)feat, (v4f*)out, n16);
    mb_update<<<B, 32, 0, stream>>>(x, tgt, feat, out, B);
}